// Decoder_19713899888784
// MI455X (gfx1250) — compile-verified
//
#include <hip/hip_runtime.h>
#include <hip/hip_bf16.h>

static constexpr int kB   = 2048;   // batch
static constexpr int kTm1 = 127;    // decoder steps / window-1
static constexpr int kE   = 128;    // ENC
static constexpr int kD   = 128;    // DEC
static constexpr int kG   = 512;    // 4*DEC gates

typedef __attribute__((ext_vector_type(2))) float v2f;
typedef __attribute__((ext_vector_type(8))) float v8f;

// f32 WMMA 16x16x4 : D = A(16x4) x B(4x16) + C, fp32 throughout.
// 8-arg pattern per probe doc: (neg_a, A, neg_b, B, c_mod, C, reuse_a, reuse_b)
__device__ __forceinline__ v8f wmma4(v2f a, v2f b, v8f c) {
  return __builtin_amdgcn_wmma_f32_16x16x4_f32(false, a, false, b, (short)0, c,
                                               false, false);
}

__device__ __forceinline__ float sigf(float x) { return 1.0f / (1.0f + __expf(-x)); }

// ---------------------------------------------------------------------------
// K0: copy initial states into ping-pong buffers
// ---------------------------------------------------------------------------
__global__ void k_init(const float* __restrict__ dn, const float* __restrict__ cn,
                       float* __restrict__ d0, float* __restrict__ c0) {
  int i = blockIdx.x * 256 + threadIdx.x;
  if (i < kB * kD) { d0[i] = dn[i]; c0[i] = cn[i]; }
}

// ---------------------------------------------------------------------------
// K1 (once): P[b,t,o] = sum_e X[b,t,e] * W1[o, 256+e] + b1[o]  -> bf16
// GEMM M = B*Tm1 (=260096, /16 exactly), K = 128, N = 128.
// grid = M/16 blocks, 8 waves/block, wave w handles N-tile w.
// ---------------------------------------------------------------------------
__global__ void k_precomputeP(const float* __restrict__ X, const float* __restrict__ W1,
                              const float* __restrict__ b1,
                              __hip_bfloat16* __restrict__ P) {
  const int mt   = blockIdx.x;
  const int nt   = threadIdx.x >> 5;      // wave id = N tile
  const int lane = threadIdx.x & 31;
  const int half = lane >> 4, l16 = lane & 15;
  const float* arow = X  + (size_t)(mt * 16 + l16) * kE  + 2 * half;
  const float* brow = W1 + (size_t)(nt * 16 + l16) * 384 + 256 + 2 * half;  // W1x slice
  v8f acc = {};
  for (int kb = 0; kb < kE; kb += 4) {
    v2f a = *(const v2f*)(arow + kb);     // A: 16x4 tile, two consecutive k per lane
    v2f b = *(const v2f*)(brow + kb);     // B[k][n] = W1x[n][k]
    acc = wmma4(a, b, acc);
  }
  const int n = nt * 16 + l16;
  const float bias = b1[n];
  const size_t base = (size_t)(mt * 16 + 8 * half) * kE + n;
#pragma unroll
  for (int r = 0; r < 8; ++r)
    P[base + (size_t)r * kE] = __float2bfloat16(acc[r] + bias);
}

// ---------------------------------------------------------------------------
// K2 (per step): Q[b,o] = W1d[o,:]·d[b,:] + W1c[o,:]·c[b,:]
// GEMM M = B, K = 256 (split), N = 128. grid = B/16, 8 waves/block.
// ---------------------------------------------------------------------------
__global__ void k_attnQ(const float* __restrict__ d, const float* __restrict__ c,
                        const float* __restrict__ W1, float* __restrict__ Q) {
  const int mt   = blockIdx.x;
  const int nt   = threadIdx.x >> 5;
  const int lane = threadIdx.x & 31;
  const int half = lane >> 4, l16 = lane & 15;
  const float* ad = d  + (size_t)(mt * 16 + l16) * kD + 2 * half;
  const float* ac = c  + (size_t)(mt * 16 + l16) * kD + 2 * half;
  const float* bd = W1 + (size_t)(nt * 16 + l16) * 384 + 0   + 2 * half;  // W1d
  const float* bc = W1 + (size_t)(nt * 16 + l16) * 384 + 128 + 2 * half;  // W1c
  v8f acc = {};
  for (int kb = 0; kb < kD; kb += 4)
    acc = wmma4(*(const v2f*)(ad + kb), *(const v2f*)(bd + kb), acc);
  for (int kb = 0; kb < kD; kb += 4)
    acc = wmma4(*(const v2f*)(ac + kb), *(const v2f*)(bc + kb), acc);
  const int n = nt * 16 + l16;
  const size_t base = (size_t)(mt * 16 + 8 * half) * kE + n;
#pragma unroll
  for (int r = 0; r < 8; ++r) Q[base + (size_t)r * kE] = acc[r];
}

// ---------------------------------------------------------------------------
// K3 (per step): scores -> softmax -> context -> y_tilde. One block per batch b.
// ---------------------------------------------------------------------------
__global__ void k_attention(const __hip_bfloat16* __restrict__ P,
                            const float* __restrict__ Q, const float* __restrict__ X,
                            const float* __restrict__ W2, const float* __restrict__ b2,
                            const float* __restrict__ fcW, const float* __restrict__ fcb,
                            const float* __restrict__ yprev, int step,
                            float* __restrict__ ctx, float* __restrict__ ytil) {
  const int b = blockIdx.x, tid = threadIdx.x;  // 128 threads
  __shared__ float sQ[128], sS[128], sR[128];
  sQ[tid] = Q[(size_t)b * kE + tid];
  sS[tid] = -3.0e38f;  // slot 127 stays -inf
  __syncthreads();

  const int wv = tid >> 5, lane = tid & 31;
  float w2v[4];
#pragma unroll
  for (int i = 0; i < 4; ++i) w2v[i] = W2[lane + 32 * i];
  const __hip_bfloat16* Pb = P + (size_t)b * kTm1 * kE;
  for (int t = wv; t < kTm1; t += 4) {
    const __hip_bfloat16* pr = Pb + (size_t)t * kE;
    float s = 0.0f;
#pragma unroll
    for (int i = 0; i < 4; ++i) {
      const int o = lane + 32 * i;
      s += w2v[i] * tanhf(__bfloat162float(pr[o]) + sQ[o]);
    }
#pragma unroll
    for (int off = 16; off; off >>= 1) s += __shfl_xor(s, off, 32);
    if (lane == 0) sS[t] = s + b2[0];
  }
  __syncthreads();

  // softmax over t = 0..126
  sR[tid] = sS[tid];
  __syncthreads();
  for (int st = 64; st; st >>= 1) {
    if (tid < st) sR[tid] = fmaxf(sR[tid], sR[tid + st]);
    __syncthreads();
  }
  const float mx = sR[0];
  const float e = (tid < kTm1) ? __expf(sS[tid] - mx) : 0.0f;
  __syncthreads();
  sR[tid] = e;
  __syncthreads();
  for (int st = 64; st; st >>= 1) {
    if (tid < st) sR[tid] += sR[tid + st];
    __syncthreads();
  }
  const float inv = 1.0f / sR[0];
  __syncthreads();
  sS[tid] = e * inv;  // beta
  __syncthreads();

  // context[o=tid] = sum_t beta[t] * X[b,t,o]   (coalesced across tid)
  const float* Xb = X + (size_t)b * kTm1 * kE + tid;
  float acc = 0.0f;
  for (int t = 0; t < kTm1; ++t) acc = fmaf(sS[t], Xb[(size_t)t * kE], acc);
  ctx[(size_t)b * kE + tid] = acc;

  // y_tilde[b] = fc_W[0:128]·ctx + fc_W[128]*y_prev[b,step] + fc_b
  __syncthreads();
  sR[tid] = fcW[tid] * acc;
  __syncthreads();
  for (int st = 64; st; st >>= 1) {
    if (tid < st) sR[tid] += sR[tid + st];
    __syncthreads();
  }
  if (tid == 0)
    ytil[b] = sR[0] + fcW[128] * yprev[(size_t)b * kTm1 + step] + fcb[0];
}

// ---------------------------------------------------------------------------
// K4 (per step): gates = d@W_hh^T + y_tilde*W_ih + b; LSTM cell update.
// grid = B/16 blocks, 8 waves; each wave does 4 of the 32 N-tiles (N=512).
// ---------------------------------------------------------------------------
__global__ void k_lstm(const float* __restrict__ dcur, const float* __restrict__ ccur,
                       const float* __restrict__ Whh, const float* __restrict__ Wih,
                       const float* __restrict__ bih, const float* __restrict__ bhh,
                       const float* __restrict__ ytil,
                       float* __restrict__ dnxt, float* __restrict__ cnxt) {
  const int mt = blockIdx.x;           // batch tile
  const int tid = threadIdx.x;         // 256
  const int wv = tid >> 5, lane = tid & 31;
  const int half = lane >> 4, l16 = lane & 15;
  __shared__ float sG[16 * kG];        // 32 KB of 320 KB LDS
  const float* arow = dcur + (size_t)(mt * 16 + l16) * kD + 2 * half;
#pragma unroll
  for (int rep = 0; rep < 4; ++rep) {
    const int nt = wv + rep * 8;
    const float* brow = Whh + (size_t)(nt * 16 + l16) * kD + 2 * half;
    v8f acc = {};
    for (int kb = 0; kb < kD; kb += 4)
      acc = wmma4(*(const v2f*)(arow + kb), *(const v2f*)(brow + kb), acc);
    const int n = nt * 16 + l16;
#pragma unroll
    for (int r = 0; r < 8; ++r) sG[(8 * half + r) * kG + n] = acc[r];
  }
  __syncthreads();

  for (int idx = tid; idx < 16 * kD; idx += 256) {
    const int bb = idx >> 7, j = idx & 127;
    const int gb = mt * 16 + bb;
    const float yt = ytil[gb];
    const float gi = sG[bb * kG + j      ] + yt * Wih[j      ] + bih[j      ] + bhh[j      ];
    const float gf = sG[bb * kG + 128 + j] + yt * Wih[128 + j] + bih[128 + j] + bhh[128 + j];
    const float gg = sG[bb * kG + 256 + j] + yt * Wih[256 + j] + bih[256 + j] + bhh[256 + j];
    const float go = sG[bb * kG + 384 + j] + yt * Wih[384 + j] + bih[384 + j] + bhh[384 + j];
    const float co = ccur[(size_t)gb * kD + j];
    const float cnv = sigf(gf) * co + sigf(gi) * tanhf(gg);
    const float dnv = sigf(go) * tanhf(cnv);
    cnxt[(size_t)gb * kD + j] = cnv;
    dnxt[(size_t)gb * kD + j] = dnv;
  }
}

// ---------------------------------------------------------------------------
// K5: y_pred = [d, ctx] @ fcf_W^T + fcf_b ; emit d and c to output.
// ---------------------------------------------------------------------------
__global__ void k_final(const float* __restrict__ dfin, const float* __restrict__ cfin,
                        const float* __restrict__ ctx, const float* __restrict__ fcfW,
                        const float* __restrict__ fcfb, float* __restrict__ out) {
  const int b = blockIdx.x, tid = threadIdx.x;  // 128
  __shared__ float sR[128];
  const float dv = dfin[(size_t)b * kD + tid];
  const float cv = cfin[(size_t)b * kD + tid];
  const float cx = ctx[(size_t)b * kE + tid];
  out[(size_t)kB + (size_t)b * kD + tid] = dv;                       // d block
  out[(size_t)kB + (size_t)kB * kD + (size_t)b * kD + tid] = cv;     // c block
  sR[tid] = fcfW[tid] * dv + fcfW[128 + tid] * cx;
  __syncthreads();
  for (int st = 64; st; st >>= 1) {
    if (tid < st) sR[tid] += sR[tid + st];
    __syncthreads();
  }
  if (tid == 0) out[b] = sR[0] + fcfb[0];
}

// ---------------------------------------------------------------------------
extern "C" void kernel_launch(void* const* d_in, const int* in_sizes, int n_in,
                              void* d_out, int out_size, void* d_ws, size_t ws_size,
                              hipStream_t stream) {
  (void)in_sizes; (void)n_in; (void)out_size; (void)ws_size;
  const float* X    = (const float*)d_in[0];
  const float* yp   = (const float*)d_in[1];
  const float* dn   = (const float*)d_in[2];
  const float* cn   = (const float*)d_in[3];
  const float* W1   = (const float*)d_in[4];
  const float* b1   = (const float*)d_in[5];
  const float* W2   = (const float*)d_in[6];
  const float* b2   = (const float*)d_in[7];
  const float* Wih  = (const float*)d_in[8];
  const float* Whh  = (const float*)d_in[9];
  const float* bih  = (const float*)d_in[10];
  const float* bhh  = (const float*)d_in[11];
  const float* fcW  = (const float*)d_in[12];
  const float* fcb  = (const float*)d_in[13];
  const float* fcfW = (const float*)d_in[14];
  const float* fcfb = (const float*)d_in[15];
  float* out = (float*)d_out;

  // workspace carve-out (~73 MB total)
  char* ws = (char*)d_ws;
  size_t off = 0;
  __hip_bfloat16* P = (__hip_bfloat16*)(ws + off); off += (size_t)kB * kTm1 * kE * 2;
  float* Q    = (float*)(ws + off); off += (size_t)kB * kE * 4;
  float* ctx  = (float*)(ws + off); off += (size_t)kB * kE * 4;
  float* ytil = (float*)(ws + off); off += (size_t)kB * 4;
  float* dbuf[2]; float* cbuf[2];
  dbuf[0] = (float*)(ws + off); off += (size_t)kB * kD * 4;
  dbuf[1] = (float*)(ws + off); off += (size_t)kB * kD * 4;
  cbuf[0] = (float*)(ws + off); off += (size_t)kB * kD * 4;
  cbuf[1] = (float*)(ws + off); off += (size_t)kB * kD * 4;

  k_init<<<(kB * kD + 255) / 256, 256, 0, stream>>>(dn, cn, dbuf[0], cbuf[0]);
  k_precomputeP<<<(kB * kTm1) / 16, 256, 0, stream>>>(X, W1, b1, P);

  for (int s = 0; s < kTm1; ++s) {
    const int cur = s & 1, nxt = cur ^ 1;
    k_attnQ<<<kB / 16, 256, 0, stream>>>(dbuf[cur], cbuf[cur], W1, Q);
    k_attention<<<kB, 128, 0, stream>>>(P, Q, X, W2, b2, fcW, fcb, yp, s, ctx, ytil);
    k_lstm<<<kB / 16, 256, 0, stream>>>(dbuf[cur], cbuf[cur], Whh, Wih, bih, bhh,
                                        ytil, dbuf[nxt], cbuf[nxt]);
  }
  // after 127 steps the live state is in buffer index 1
  k_final<<<kB, 128, 0, stream>>>(dbuf[1], cbuf[1], ctx, fcfW, fcfb, out);
}